// conv_in_seq_direction_moment_knn_833223655548
// MI455X (gfx1250) — compile-verified
//
#include <hip/hip_runtime.h>
#include <math.h>

// ---------------------------------------------------------------------------
// Geometry constants (match reference)
// ---------------------------------------------------------------------------
#define BATCH  8
#define NPTS   4096
#define KNN    10
#define GROUPS 4
#define EPSV   1e-5f

typedef __attribute__((ext_vector_type(2))) float v2f;
typedef __attribute__((ext_vector_type(8))) float v8f;

static __device__ __forceinline__ float gelu_f(float v) {
    const float c = 0.7978845608028654f; // sqrt(2/pi)
    float u = c * (v + 0.044715f * v * v * v);
    return 0.5f * v * (1.0f + tanhf(u));
}

// ---------------------------------------------------------------------------
// Kernel 1: kNN indices (top-10 of pd = 2*inner - |q|^2 - |p|^2) per 3-ch set.
// grid = (N/256, B, 2), block = 256. LDS-tiled candidates (+precomputed |p|^2),
// register-resident top-10 via fully-unrolled swap-chain insertion.
// ---------------------------------------------------------------------------
__global__ __launch_bounds__(256)
void knn_kernel(const float* __restrict__ x, int* __restrict__ idx) {
    __shared__ float sp[256 * 4];
    const int tid = threadIdx.x;
    const int n   = blockIdx.x * 256 + tid;
    const int b   = blockIdx.y;
    const int s   = blockIdx.z;
    const float* xb = x + ((size_t)b * 6 + s * 3) * NPTS;

    const float q0 = xb[n], q1 = xb[NPTS + n], q2 = xb[2 * NPTS + n];
    const float qq = q0 * q0 + q1 * q1 + q2 * q2;

    float best[KNN];
    int   bi[KNN];
#pragma unroll
    for (int j = 0; j < KNN; ++j) { best[j] = -3.0e38f; bi[j] = 0; }

    for (int t0 = 0; t0 < NPTS; t0 += 256) {
        const float p0 = xb[t0 + tid];
        const float p1 = xb[NPTS + t0 + tid];
        const float p2 = xb[2 * NPTS + t0 + tid];
        sp[tid]       = p0;
        sp[256 + tid] = p1;
        sp[512 + tid] = p2;
        sp[768 + tid] = p0 * p0 + p1 * p1 + p2 * p2;   // |p|^2 once per tile
        __syncthreads();
        for (int m = 0; m < 256; ++m) {
            const float inner = q0 * sp[m] + q1 * sp[256 + m] + q2 * sp[512 + m];
            const float pd    = 2.0f * inner - qq - sp[768 + m];
            if (pd > best[KNN - 1]) {
                float v = pd; int id = t0 + m;
#pragma unroll
                for (int j = 0; j < KNN; ++j) {
                    if (v > best[j]) {
                        float tv = best[j]; int ti = bi[j];
                        best[j] = v; bi[j] = id;
                        v = tv; id = ti;
                    }
                }
            }
        }
        __syncthreads();
    }
    int* out = idx + (((size_t)s * BATCH + b) * NPTS + n) * KNN;
#pragma unroll
    for (int j = 0; j < KNN; ++j) out[j] = bi[j];
}

// ---------------------------------------------------------------------------
// Kernel 2: edge-conv (6 -> 8) with mean-over-k folded in:
//   h[o][n] = b[o] + W[o,:] @ [mean_nbr - center, center]
// ---------------------------------------------------------------------------
__global__ __launch_bounds__(256)
void edgeconv_kernel(const float* __restrict__ x, const int* __restrict__ idx,
                     const float* __restrict__ Wd, const float* __restrict__ bd,
                     const float* __restrict__ Wm, const float* __restrict__ bm,
                     float* __restrict__ h) {
    const int n = blockIdx.x * 256 + threadIdx.x;
    const int b = blockIdx.y;
    const int s = blockIdx.z;
    const float* xb = x + ((size_t)b * 6 + s * 3) * NPTS;
    const int* id = idx + (((size_t)s * BATCH + b) * NPTS + n) * KNN;

    const float c0 = xb[n], c1 = xb[NPTS + n], c2 = xb[2 * NPTS + n];
    float m0 = 0.f, m1 = 0.f, m2 = 0.f;
#pragma unroll
    for (int j = 0; j < KNN; ++j) {
        const int m = id[j];
        m0 += xb[m]; m1 += xb[NPTS + m]; m2 += xb[2 * NPTS + m];
    }
    const float inv = 1.0f / (float)KNN;
    const float f[6] = { m0 * inv - c0, m1 * inv - c1, m2 * inv - c2, c0, c1, c2 };

    const float* W  = s ? Wm : Wd;
    const float* bb = s ? bm : bd;
    float* hb = h + ((size_t)s * BATCH + b) * 8 * NPTS;
#pragma unroll
    for (int o = 0; o < 8; ++o) {
        float acc = bb[o];
#pragma unroll
        for (int c = 0; c < 6; ++c) acc += W[o * 6 + c] * f[c];
        hb[(size_t)o * NPTS + n] = acc;
    }
}

// ---------------------------------------------------------------------------
// Kernel 3: 1x1 conv as GEMM on the matrix pipe (V_WMMA_F32_16X16X4_F32).
//   Y[b, o_off+o, n] = bias[o] + sum_c W[o,c] * X[b,c,n]
// Template C -> fully-unrolled K loop, no runtime division anywhere.
// grid = (tilesN/(8*TN), O/16, B), block = 256 (8 waves).
// Each wave computes TN adjacent 16x16 tiles sharing one A fragment per
// K-step (4 WMMAs per 2 weight loads). All waves active -> EXEC all-1s.
// Fragment layouts per CDNA5 ISA 7.12.2 (32-bit A 16x4 / B 4x16 / C 16x16).
// ---------------------------------------------------------------------------
template<int C, int TN>
__global__ __launch_bounds__(256)
void conv_wmma_kernel(const float* __restrict__ X, const float* __restrict__ W,
                      const float* __restrict__ bias, float* __restrict__ Y,
                      int o_off, int OC_total) {
    const int waveId = threadIdx.x >> 5;
    const int lane   = threadIdx.x & 31;
    const int half   = lane >> 4;            // 0: lanes 0-15, 1: lanes 16-31
    const int l      = lane & 15;

    const int mBase  = blockIdx.y * 16;
    const int nBase  = (blockIdx.x * 8 + waveId) * (TN * 16);
    const int b      = blockIdx.z;

    const float* Xb = X + (size_t)b * C * NPTS;

    v8f acc[TN];
#pragma unroll
    for (int t = 0; t < TN; ++t) acc[t] = (v8f){};

#pragma unroll
    for (int k = 0; k < C; k += 4) {
        // A (16x4): lanes 0-15 hold K=k,k+1; lanes 16-31 hold K=k+2,k+3 (row M=l)
        const float* wrow = W + (mBase + l) * C + k + half * 2;
        v2f a; a.x = wrow[0]; a.y = wrow[1];
        // B (4x16): lanes 0-15 hold K=k,k+1; lanes 16-31 hold K=k+2,k+3 (col N=l)
        const float* xrow = Xb + (size_t)(k + half * 2) * NPTS + nBase + l;
#pragma unroll
        for (int t = 0; t < TN; ++t) {
            v2f bf; bf.x = xrow[t * 16]; bf.y = xrow[t * 16 + NPTS];
            acc[t] = __builtin_amdgcn_wmma_f32_16x16x4_f32(
                /*neg_a=*/false, a, /*neg_b=*/false, bf,
                /*c_mod=*/(short)0, acc[t], /*reuse_a=*/false, /*reuse_b=*/false);
        }
    }

    // D (16x16): VGPR r -> row mBase + r (+8 for lanes 16-31), col nBase + l
    float bval[8];
#pragma unroll
    for (int r = 0; r < 8; ++r) bval[r] = bias[mBase + r + half * 8];

    float* Yb = Y + (size_t)b * OC_total * NPTS;
#pragma unroll
    for (int t = 0; t < TN; ++t) {
#pragma unroll
        for (int r = 0; r < 8; ++r) {
            const int row = mBase + r + half * 8;
            Yb[(size_t)(row + o_off) * NPTS + nBase + t * 16 + l] = acc[t][r] + bval[r];
        }
    }
}

// ---------------------------------------------------------------------------
// Kernel 4: GroupNorm statistics. grid = (GROUPS, B), block = 256.
// float4 (b128) grid-stride reads, LDS tree reduction.
// ---------------------------------------------------------------------------
template<int C>
__global__ __launch_bounds__(256)
void gn_stats_kernel(const float* __restrict__ a, float* __restrict__ stats) {
    const int g = blockIdx.x, b = blockIdx.y;
    constexpr int Cg    = C / GROUPS;
    constexpr int total = Cg * NPTS;
    const float4* ab4 = reinterpret_cast<const float4*>(a + ((size_t)b * C + (size_t)g * Cg) * NPTS);

    float s = 0.f, s2 = 0.f;
    for (int i = threadIdx.x; i < total / 4; i += 256) {
        const float4 v = ab4[i];
        s  += v.x + v.y + v.z + v.w;
        s2 += v.x * v.x + v.y * v.y + v.z * v.z + v.w * v.w;
    }
    __shared__ float sh[512];
    sh[threadIdx.x]       = s;
    sh[256 + threadIdx.x] = s2;
    __syncthreads();
    for (int st = 128; st > 0; st >>= 1) {
        if (threadIdx.x < st) {
            sh[threadIdx.x]       += sh[threadIdx.x + st];
            sh[256 + threadIdx.x] += sh[256 + threadIdx.x + st];
        }
        __syncthreads();
    }
    if (threadIdx.x == 0) {
        const float mean = sh[0] / (float)total;
        const float var  = sh[256] / (float)total - mean * mean;
        stats[(b * GROUPS + g) * 2]     = mean;
        stats[(b * GROUPS + g) * 2 + 1] = rsqrtf(var + EPSV);
    }
}

// ---------------------------------------------------------------------------
// Kernel 5: fused GroupNorm apply + affine + GELU (in place).
// Template C (power of two) -> index math is shifts/masks only.
// ---------------------------------------------------------------------------
template<int C>
__global__ __launch_bounds__(256)
void gn_gelu_kernel(float* __restrict__ a, const float* __restrict__ stats,
                    const float* __restrict__ w, const float* __restrict__ bias) {
    const size_t i = (size_t)blockIdx.x * 256 + threadIdx.x;
    const int cdiv = (int)(i / NPTS);
    const int c    = cdiv % C;
    const int b    = cdiv / C;
    const int g    = c / (C / GROUPS);
    const float mean = stats[(b * GROUPS + g) * 2];
    const float rstd = stats[(b * GROUPS + g) * 2 + 1];
    const float v = (a[i] - mean) * rstd * w[c] + bias[c];
    a[i] = gelu_f(v);
}

// ---------------------------------------------------------------------------
// Host-side templated launch helpers
// ---------------------------------------------------------------------------
template<int C>
static void launch_conv(const float* X, const float* W, const float* b, float* Y,
                        int O, int o_off, int OCt, hipStream_t stream) {
    constexpr int TN = 4;
    dim3 grid((NPTS / 16) / (8 * TN), O / 16, BATCH);  // (8, O/16, 8)
    conv_wmma_kernel<C, TN><<<grid, 256, 0, stream>>>(X, W, b, Y, o_off, OCt);
}

template<int C>
static void launch_gnorm(float* A, float* st, const float* gw, const float* gb,
                         hipStream_t stream) {
    gn_stats_kernel<C><<<dim3(GROUPS, BATCH), 256, 0, stream>>>(A, st);
    gn_gelu_kernel<C><<<dim3((BATCH * C * NPTS) / 256), 256, 0, stream>>>(A, st, gw, gb);
}

// ---------------------------------------------------------------------------
// Entry point
// ---------------------------------------------------------------------------
extern "C" void kernel_launch(void* const* d_in, const int* in_sizes, int n_in,
                              void* d_out, int out_size, void* d_ws, size_t ws_size,
                              hipStream_t stream) {
    // ---- unpack inputs (handles insertion-order or sorted-pytree flattening) --
    auto F = [&](int i) { return (const float*)d_in[i]; };
    const float *x, *cdW, *cdb, *cmW, *cmb;
    const float *dW1,*db1,*dg1w,*dg1b,*dW2,*db2,*dg2w,*dg2b,*dW3,*db3;
    const float *mW1,*mb1,*mg1w,*mg1b,*mW2,*mb2,*mg2w,*mg2b,*mW3,*mb3;
    const float *MW1,*Mb1,*Mg1w,*Mg1b,*MW2,*Mb2;

    if (n_in > 0 && in_sizes[0] == BATCH * 6 * NPTS) {
        // insertion order: x, conv_dir{W,b}, conv_mom{W,b},
        // mlp_dir{W1,b1,g1w,g1b,W2,b2,g2w,g2b,W3,b3}, mlp_mom{...},
        // mlp_merged{W1,b1,g1w,g1b,W2,b2}
        int i = 0;
        x = F(i++); cdW = F(i++); cdb = F(i++); cmW = F(i++); cmb = F(i++);
        dW1=F(i++); db1=F(i++); dg1w=F(i++); dg1b=F(i++); dW2=F(i++); db2=F(i++); dg2w=F(i++); dg2b=F(i++); dW3=F(i++); db3=F(i++);
        mW1=F(i++); mb1=F(i++); mg1w=F(i++); mg1b=F(i++); mW2=F(i++); mb2=F(i++); mg2w=F(i++); mg2b=F(i++); mW3=F(i++); mb3=F(i++);
        MW1=F(i++); Mb1=F(i++); Mg1w=F(i++); Mg1b=F(i++); MW2=F(i++); Mb2=F(i++);
    } else {
        // sorted-key pytree order: params first (keys sorted at every level), x last
        cdW=F(0); cdb=F(1); cmW=F(2); cmb=F(3);
        dW1=F(4); dW2=F(5); dW3=F(6); db1=F(7); db2=F(8); db3=F(9);
        dg1b=F(10); dg1w=F(11); dg2b=F(12); dg2w=F(13);
        MW1=F(14); MW2=F(15); Mb1=F(16); Mb2=F(17); Mg1b=F(18); Mg1w=F(19);
        mW1=F(20); mW2=F(21); mW3=F(22); mb1=F(23); mb2=F(24); mb3=F(25);
        mg1b=F(26); mg1w=F(27); mg2b=F(28); mg2w=F(29);
        x = F(30);
    }

    // ---- workspace layout (bytes) ----
    char* ws = (char*)d_ws;
    size_t off = 0;
    int*   idx = (int*)(ws + off);   off += (size_t)BATCH * 2 * NPTS * KNN * 4;     // 2.62 MB
    float* h   = (float*)(ws + off); off += (size_t)2 * BATCH * 8 * NPTS * 4;       // 2.10 MB
    float* t1  = (float*)(ws + off); off += (size_t)BATCH * 16 * NPTS * 4;          // 2.10 MB
    float* t2  = (float*)(ws + off); off += (size_t)BATCH * 32 * NPTS * 4;          // 4.19 MB
    float* xc  = (float*)(ws + off); off += (size_t)BATCH * 128 * NPTS * 4;         // 16.8 MB
    float* tm  = (float*)(ws + off); off += (size_t)BATCH * 128 * NPTS * 4;         // 16.8 MB
    float* st  = (float*)(ws + off); off += (size_t)BATCH * GROUPS * 2 * 4;
    (void)ws_size;

    float* out = (float*)d_out;      // (8, 128, 4096) f32

    // 1) kNN for both channel sets
    knn_kernel<<<dim3(NPTS / 256, BATCH, 2), 256, 0, stream>>>(x, idx);

    // 2) edge conv (6->8), mean over k folded in
    edgeconv_kernel<<<dim3(NPTS / 256, BATCH, 2), 256, 0, stream>>>(x, idx, cdW, cdb, cmW, cmb, h);

    // 3) direction branch: 8 -> 16 -> 32 -> 64  (writes xc channels [0,64))
    {
        const float* hdir = h;                       // set 0
        launch_conv<8 >(hdir, dW1, db1, t1, 16, 0, 16, stream);
        launch_gnorm<16>(t1, st, dg1w, dg1b, stream);
        launch_conv<16>(t1, dW2, db2, t2, 32, 0, 32, stream);
        launch_gnorm<32>(t2, st, dg2w, dg2b, stream);
        launch_conv<32>(t2, dW3, db3, xc, 64, 0, 128, stream);
    }
    // 4) moment branch: writes xc channels [64,128)
    {
        const float* hmom = h + (size_t)BATCH * 8 * NPTS;  // set 1
        launch_conv<8 >(hmom, mW1, mb1, t1, 16, 0, 16, stream);
        launch_gnorm<16>(t1, st, mg1w, mg1b, stream);
        launch_conv<16>(t1, mW2, mb2, t2, 32, 0, 32, stream);
        launch_gnorm<32>(t2, st, mg2w, mg2b, stream);
        launch_conv<32>(t2, mW3, mb3, xc, 64, 64, 128, stream);
    }
    // 5) merged head: 128 -> 128 (GN+GELU) -> 128
    launch_conv<128>(xc, MW1, Mb1, tm, 128, 0, 128, stream);
    launch_gnorm<128>(tm, st, Mg1w, Mg1b, stream);
    launch_conv<128>(tm, MW2, Mb2, out, 128, 0, 128, stream);

    (void)out_size;
}